// GCN_model_82351702934073
// MI455X (gfx1250) — compile-verified
//
#include <hip/hip_runtime.h>
#include <hip/hip_bf16.h>

typedef __attribute__((ext_vector_type(2))) float v2f;
typedef __attribute__((ext_vector_type(8))) float v8f;

static constexpr int   BN   = 8;
static constexpr int   NPER = 32768;
static constexpr int   NN   = BN * NPER;            // 262144 nodes
static constexpr long long EE = (long long)NN * 8;  // 2097152 edges
static constexpr int   DD   = 64;
static constexpr int   KK   = 16;
static constexpr long long MC_OFF = 8LL + (long long)NN * KK;     // 4194312
static constexpr long long O1_OFF = MC_OFF + 1;                   // 4194313

#if defined(__AMDGCN__) && __has_builtin(__builtin_amdgcn_global_load_async_to_lds_b128) && __has_builtin(__builtin_amdgcn_s_wait_asynccnt)
#define HAVE_ASYNC_LDS 1
typedef __attribute__((__vector_size__(4 * sizeof(int)))) int v4i;   // matches builtin param
#else
#define HAVE_ASYNC_LDS 0
#endif

// ---------------------------------------------------------------- fills
__global__ void fill_f32(float* __restrict__ p, float v, long long n) {
    long long i = (long long)blockIdx.x * blockDim.x + threadIdx.x;
    long long stride = (long long)gridDim.x * blockDim.x;
    for (; i < n; i += stride) p[i] = v;
}

__global__ void fill_i32(int* __restrict__ p, int v, long long n) {
    long long i = (long long)blockIdx.x * blockDim.x + threadIdx.x;
    long long stride = (long long)gridDim.x * blockDim.x;
    for (; i < n; i += stride) p[i] = v;
}

// ---------------------------------------------------------------- degrees
__global__ void count_kernel(const int* __restrict__ src, const int* __restrict__ dst,
                             int* __restrict__ cnt, float* __restrict__ dega) {
    long long e = (long long)blockIdx.x * 256 + threadIdx.x;
    atomicAdd(&cnt[dst[e]], 1);        // in-degree (without self loop)
    atomicAdd(&dega[src[e]], 1.0f);    // out-degree (deg_a in reference)
}

__global__ void dis_kernel(const int* __restrict__ cnt, float* __restrict__ dis) {
    int i = blockIdx.x * 256 + threadIdx.x;
    dis[i] = rsqrtf(1.0f + (float)cnt[i]);   // +1 self loop
}

// ---------------------------------------------------------------- scan (exclusive)
__global__ __launch_bounds__(256) void scan_block(const int* __restrict__ in,
                                                  int* __restrict__ out,
                                                  int* __restrict__ sums) {
    __shared__ int sh[256];
    int t = threadIdx.x;
    int g = blockIdx.x * 256 + t;
    int v = in[g];
    sh[t] = v;
    __syncthreads();
    for (int off = 1; off < 256; off <<= 1) {
        int add = (t >= off) ? sh[t - off] : 0;
        __syncthreads();
        sh[t] += add;
        __syncthreads();
    }
    out[g] = sh[t] - v;                       // exclusive
    if (t == 255) sums[blockIdx.x] = sh[255]; // block total
}

__global__ __launch_bounds__(1024) void scan_sums(int* __restrict__ sums) {
    __shared__ int sh[1024];
    int t = threadIdx.x;
    int v = sums[t];
    sh[t] = v;
    __syncthreads();
    for (int off = 1; off < 1024; off <<= 1) {
        int add = (t >= off) ? sh[t - off] : 0;
        __syncthreads();
        sh[t] += add;
        __syncthreads();
    }
    sums[t] = sh[t] - v;                      // exclusive
}

__global__ void scan_add(int* __restrict__ out, const int* __restrict__ sums) {
    int g = blockIdx.x * 256 + threadIdx.x;
    out[g] += sums[blockIdx.x];
}

// ---------------------------------------------------------------- CSR fill
__global__ void csr_fill(const int* __restrict__ src, const int* __restrict__ dst,
                         const int* __restrict__ offs, int* __restrict__ cursor,
                         int* __restrict__ csrSrc) {
    long long e = (long long)blockIdx.x * 256 + threadIdx.x;
    int d = dst[e];
    int pos = offs[d] + atomicAdd(&cursor[d], 1);
    csrSrc[pos] = src[e];
}

// ---------------------------------------------------------------- embedding
__global__ void embed_kernel(const int* __restrict__ x, const float* __restrict__ emb,
                             float* __restrict__ h) {
    long long t = (long long)blockIdx.x * 256 + threadIdx.x;
    int node = (int)(t >> 6);
    int d    = (int)(t & 63);
    h[t] = emb[x[node] * DD + d];
}

// ---------------------------------------------------------------- WMMA GEMM
// O[N x NC] = A[N x 64] @ W[64 x NC] (+ bias). One wave = one 16x16 C tile,
// 16 steps of V_WMMA_F32_16X16X4_F32 (fp32 precision). W staged in LDS via
// the CDNA5 async global->LDS path when available (ASYNCcnt tracked).
__global__ __launch_bounds__(256) void gemm_wmma(const float* __restrict__ A,
                                                 const float* __restrict__ W,
                                                 const float* __restrict__ bias,
                                                 float* __restrict__ O,
                                                 int NC, int tcn) {
    __shared__ float wlds[DD * DD];          // up to 16 KB (64x64)
    const int nchunks = (DD * NC) / 4;       // float4 chunks of W
    for (int j = threadIdx.x; j < nchunks; j += 256) {
#if HAVE_ASYNC_LDS
        __builtin_amdgcn_global_load_async_to_lds_b128(
            (v4i*)(W + j * 4), (v4i*)(wlds + j * 4), 0, 0);
#else
        ((float4*)wlds)[j] = ((const float4*)W)[j];
#endif
    }
#if HAVE_ASYNC_LDS
    __builtin_amdgcn_s_wait_asynccnt(0);
#endif
    __syncthreads();

    int wave = (int)((blockIdx.x * 256 + threadIdx.x) >> 5);
    int lane = threadIdx.x & 31;
    int tr = wave / tcn;
    int tc = wave - tr * tcn;
    int r0 = tr * 16, c0 = tc * 16;
    int half = lane >> 4;       // 0: K pair {0,1}; 1: K pair {2,3}
    int lm   = lane & 15;       // A: row in tile; B/C: col in tile

    const float* arow = A + (size_t)(r0 + lm) * DD;
    v8f c = {};
    #pragma unroll
    for (int kb = 0; kb < DD; kb += 4) {
        int ka = kb + 2 * half;
        v2f a, b;
        a.x = arow[ka];
        a.y = arow[ka + 1];
        b.x = wlds[ka * NC + c0 + lm];
        b.y = wlds[(ka + 1) * NC + c0 + lm];
        c = __builtin_amdgcn_wmma_f32_16x16x4_f32(
                /*neg_a=*/false, a, /*neg_b=*/false, b,
                /*c_mod=*/(short)0, c, /*reuse_a=*/false, /*reuse_b=*/false);
    }
    #pragma unroll
    for (int v = 0; v < 8; ++v) {
        int row = r0 + 8 * half + v;    // C/D: lanes 0-15 M=v, lanes 16-31 M=8+v
        int col = c0 + lm;
        float val = c[v] + (bias ? bias[col] : 0.0f);
        O[(size_t)row * NC + col] = val;
    }
}

// ---------------------------------------------------------------- gather agg
// One wave per node: registers accumulate 64 features over in-edges +
// self-loop; fused bias + ReLU. No float atomics at all.
__global__ __launch_bounds__(256) void gather_agg(const int* __restrict__ offs,
                                                  const int* __restrict__ cnt,
                                                  const int* __restrict__ csrSrc,
                                                  const float* __restrict__ dis,
                                                  const float* __restrict__ M,
                                                  const float* __restrict__ bias,
                                                  float* __restrict__ O) {
    int i    = (int)((blockIdx.x * 256 + threadIdx.x) >> 5);   // node id
    int lane = threadIdx.x & 31;
    float di = dis[i];
    float wself = di * di;                    // self-loop norm = 1/deg
    float acc0 = M[(size_t)i * DD + lane]      * wself;
    float acc1 = M[(size_t)i * DD + 32 + lane] * wself;
    int beg = offs[i];
    int n   = cnt[i];
    for (int j = 0; j < n; ++j) {
        int s = csrSrc[beg + j];
        float w = dis[s] * di;
        acc0 += M[(size_t)s * DD + lane]      * w;
        acc1 += M[(size_t)s * DD + 32 + lane] * w;
    }
    float v0 = acc0 + bias[lane];
    float v1 = acc1 + bias[32 + lane];
    O[(size_t)i * DD + lane]      = v0 > 0.0f ? v0 : 0.0f;
    O[(size_t)i * DD + 32 + lane] = v1 > 0.0f ? v1 : 0.0f;
}

// ---------------------------------------------------------------- softmax
__global__ void node_softmax(const float* __restrict__ S, const int* __restrict__ batch,
                             const float* __restrict__ dega, float* __restrict__ outSB,
                             float* __restrict__ den, float* __restrict__ xp) {
    int i = blockIdx.x * 256 + threadIdx.x;
    const float* si = S + (size_t)i * KK;
    float p[KK];
    float mx = si[0];
    #pragma unroll
    for (int k = 1; k < KK; ++k) mx = fmaxf(mx, si[k]);
    float sum = 0.0f;
    #pragma unroll
    for (int k = 0; k < KK; ++k) { p[k] = expf(si[k] - mx); sum += p[k]; }
    float inv = 1.0f / sum;
    int b = batch[i];
    float* o = outSB + (size_t)i * KK;
    float ssq = 0.0f;
    #pragma unroll
    for (int k = 0; k < KK; ++k) {
        float v = p[k] * inv;
        o[k] = v;
        ssq += v * v;
        atomicAdd(&xp[b * KK + k], v);
    }
    atomicAdd(&den[b], dega[i] * ssq);
}

// ---------------------------------------------------------------- mincut num
__global__ void edge_num(const int* __restrict__ src, const int* __restrict__ dst,
                         const int* __restrict__ batch, const float* __restrict__ sb,
                         float* __restrict__ num) {
    long long e = (long long)blockIdx.x * 256 + threadIdx.x;
    int s = src[e], d = dst[e];
    const float* ps = sb + (size_t)s * KK;
    const float* pd = sb + (size_t)d * KK;
    float dot = 0.0f;
    #pragma unroll
    for (int k = 0; k < KK; ++k) dot += ps[k] * pd[k];
    atomicAdd(&num[batch[s]], dot);
}

// ---------------------------------------------------------------- SS = s^T s
__global__ __launch_bounds__(256) void ss_kernel(const float* __restrict__ sb,
                                                 float* __restrict__ SS) {
    __shared__ float sh[128 * KK];
    int b     = blockIdx.x >> 8;              // 256 chunks per batch
    int chunk = blockIdx.x & 255;
    const float* base = sb + ((size_t)b * NPER + (size_t)chunk * 128) * KK;
    for (int j = threadIdx.x; j < 128 * KK; j += 256) sh[j] = base[j];
    __syncthreads();
    int k = threadIdx.x >> 4, l = threadIdx.x & 15;
    float acc = 0.0f;
    #pragma unroll 4
    for (int n = 0; n < 128; ++n) acc += sh[n * KK + k] * sh[n * KK + l];
    atomicAdd(&SS[b * 256 + threadIdx.x], acc);
}

// ---------------------------------------------------------------- finalize
__global__ __launch_bounds__(256) void finalize(const float* __restrict__ num,
                                                const float* __restrict__ den,
                                                const float* __restrict__ SS,
                                                const float* __restrict__ xp,
                                                const float* __restrict__ ncells,
                                                const float* __restrict__ Wp,
                                                const float* __restrict__ bp,
                                                float* __restrict__ out) {
    __shared__ float red[256];
    int t = threadIdx.x;
    float o1acc = 0.0f;
    for (int b = 0; b < BN; ++b) {
        float v = SS[b * 256 + t];
        red[t] = v * v;
        __syncthreads();
        for (int s2 = 128; s2 > 0; s2 >>= 1) {
            if (t < s2) red[t] += red[t + s2];
            __syncthreads();
        }
        float ssn = sqrtf(red[0]);
        __syncthreads();
        float diag = ((t & 15) == (t >> 4)) ? 0.25f : 0.0f;   // I/sqrt(16)
        float diff = v / ssn - diag;
        red[t] = diff * diff;
        __syncthreads();
        for (int s2 = 128; s2 > 0; s2 >>= 1) {
            if (t < s2) red[t] += red[t + s2];
            __syncthreads();
        }
        if (t == 0) o1acc += sqrtf(red[0]);
        __syncthreads();
    }
    if (t == 0) {
        out[O1_OFF] = o1acc / (float)BN;
        float mc = 0.0f;
        for (int b = 0; b < BN; ++b) mc += -num[b] / den[b];
        out[MC_OFF] = mc / (float)BN;
    }
    if (t < BN) {
        float acc = bp[0];
        float invn = 1.0f / ncells[t];
        for (int k = 0; k < KK; ++k) acc += (xp[t * KK + k] * invn) * Wp[k];
        out[t] = acc;
    }
}

// ---------------------------------------------------------------- launcher
extern "C" void kernel_launch(void* const* d_in, const int* in_sizes, int n_in,
                              void* d_out, int out_size, void* d_ws, size_t ws_size,
                              hipStream_t stream) {
    (void)in_sizes; (void)n_in; (void)out_size; (void)ws_size;
    const int*   x      = (const int*)d_in[0];
    const int*   ei     = (const int*)d_in[1];
    const int*   src    = ei;
    const int*   dst    = ei + EE;
    const int*   batch  = (const int*)d_in[2];
    const float* ncells = (const float*)d_in[3];
    const float* emb    = (const float*)d_in[4];
    const float* W2     = (const float*)d_in[5];
    const float* b2     = (const float*)d_in[6];
    const float* W3     = (const float*)d_in[7];
    const float* b3     = (const float*)d_in[8];
    const float* W1     = (const float*)d_in[9];
    const float* b1     = (const float*)d_in[10];
    const float* Wp     = (const float*)d_in[11];
    const float* bp     = (const float*)d_in[12];
    float* out = (float*)d_out;

    float* ws = (float*)d_ws;
    const size_t N64 = (size_t)NN * DD;
    float* bufA = ws;                 // N x 64
    float* bufB = ws + N64;           // N x 64
    float* bufC = ws + 2 * N64;       // N x 64 (later N x 16 logits)
    float* dis  = ws + 3 * N64;       // N
    float* dega = dis + NN;           // N
    float* num  = dega + NN;          // 8
    float* den  = num + BN;           // 8
    float* SS   = den + BN;           // 8*256
    float* xp   = SS + BN * 256;      // 8*16
    int*   icnt   = (int*)(xp + BN * KK);   // N
    int*   offs   = icnt + NN;              // N
    int*   cursor = offs + NN;              // N
    int*   csrSrc = cursor + NN;            // E
    int*   bsums  = csrSrc + EE;            // 1024

    const int nodeBlocks = NN / 256;             // 1024
    const int edgeBlocks = (int)(EE / 256);      // 8192
    const int n64Blocks  = (int)(N64 / 256);     // 65536

    // ---- CSR build + degrees
    fill_i32<<<1024, 256, 0, stream>>>(icnt, 0, (long long)NN);
    fill_f32<<<1024, 256, 0, stream>>>(dega, 0.0f, (long long)NN);
    fill_f32<<<16, 256, 0, stream>>>(num, 0.0f, (long long)(BN + BN + BN * 256 + BN * KK));
    count_kernel<<<edgeBlocks, 256, 0, stream>>>(src, dst, icnt, dega);
    dis_kernel<<<nodeBlocks, 256, 0, stream>>>(icnt, dis);
    scan_block<<<nodeBlocks, 256, 0, stream>>>(icnt, offs, bsums);
    scan_sums<<<1, 1024, 0, stream>>>(bsums);
    scan_add<<<nodeBlocks, 256, 0, stream>>>(offs, bsums);
    fill_i32<<<1024, 256, 0, stream>>>(cursor, 0, (long long)NN);
    csr_fill<<<edgeBlocks, 256, 0, stream>>>(src, dst, offs, cursor, csrSrc);

    // ---- h = emb[x]
    embed_kernel<<<n64Blocks, 256, 0, stream>>>(x, emb, bufA);

    // ---- layer 1: m = h @ W2 ; gather-agg (+b2, relu)
    gemm_wmma<<<(NN / 16) * 4 / 8, 256, 0, stream>>>(bufA, W2, nullptr, bufB, 64, 4);
    gather_agg<<<NN * 32 / 256, 256, 0, stream>>>(offs, icnt, csrSrc, dis, bufB, b2, bufC);

    // ---- layer 2: m = out @ W3 ; gather-agg (+b3, relu)
    gemm_wmma<<<(NN / 16) * 4 / 8, 256, 0, stream>>>(bufC, W3, nullptr, bufA, 64, 4);
    gather_agg<<<NN * 32 / 256, 256, 0, stream>>>(offs, icnt, csrSrc, dis, bufA, b3, bufB);

    // ---- s = out @ W1 + b1 -> softmax -> s_b (into d_out), den, xp
    gemm_wmma<<<(NN / 16) / 8, 256, 0, stream>>>(bufB, W1, b1, bufC, 16, 1);
    node_softmax<<<nodeBlocks, 256, 0, stream>>>(bufC, batch, dega, out + 8, den, xp);

    // ---- mincut numerator + SS + scalars/pred
    edge_num<<<edgeBlocks, 256, 0, stream>>>(src, dst, batch, out + 8, num);
    ss_kernel<<<BN * 256, 256, 0, stream>>>(out + 8, SS);
    finalize<<<1, 256, 0, stream>>>(num, den, SS, xp, ncells, Wp, bp, out);
}